// ScaledDotProductAttention_5059471475343
// MI455X (gfx1250) — compile-verified
//
#include <hip/hip_runtime.h>
#include <hip/hip_bf16.h>

typedef float v2f __attribute__((ext_vector_type(2)));
typedef float v8f __attribute__((ext_vector_type(8)));

#define EMB 4096
#define BM 128
#define BN 128
#define BK 16
#define SA 20    // LDS row stride (floats) for [row][k] tiles: 16 + 4 pad
#define SBN 132  // LDS row stride (floats) for [k][n] tile: 128 + 4 pad

// C[M,N] = alpha * A @ (BT ? B^T : B), all row-major fp32, square EMB x EMB.
// BT=true : C[i,j] = sum_k A[i,k] * B[j,k]   (scores = Wq @ Wk^T)
// BT=false: C[i,j] = sum_k A[i,k] * B[k,j]   (sftmax = attn @ Wv)
template <bool BT>
__global__ __launch_bounds__(128) void gemm_wmma_f32(const float* __restrict__ A,
                                                     const float* __restrict__ B,
                                                     float* __restrict__ C,
                                                     float alpha) {
  // double-buffered staging
  __shared__ float As[2][BM * SA];
  constexpr int BSZ = BT ? (BN * SA) : (BK * SBN);
  __shared__ float Bs[2][BSZ];

  const int tid   = threadIdx.x;
  const int lane  = tid & 31;
  const int wave  = tid >> 5;
  const int lmod  = lane & 15;   // M (A) / N (B,C) index within 16
  const int lhalf = lane >> 4;   // selects K pair (frags) / M half (C)

  const int blockM = blockIdx.y * BM;
  const int blockN = blockIdx.x * BN;
  const int waveM  = (wave >> 1) * 64;   // 2x2 wave grid, 64x64 per wave
  const int waveN  = (wave & 1) * 64;

  v8f acc[4][4] = {};

  // A-tile loader: 128 rows x 16 cols, 128 threads, 4 float4 each
  const int ar = tid >> 2;        // 0..31 -> rows ar + 32*i
  const int ac = (tid & 3) * 4;   // 0,4,8,12
  // B-tile loader for the non-transposed (16 x 128) case
  const int br = tid >> 3;        // 0..15
  const int bc = (tid & 7) * 4;   // 0..28 -> cols bc + 32*i (i=0..3)

  float4 rega[4], regb[4];

  auto load_tile = [&](int k0) {
#pragma unroll
    for (int i = 0; i < 4; ++i)
      rega[i] = *(const float4*)&A[(size_t)(blockM + ar + 32 * i) * EMB + k0 + ac];
    if constexpr (BT) {
#pragma unroll
      for (int i = 0; i < 4; ++i)
        regb[i] = *(const float4*)&B[(size_t)(blockN + ar + 32 * i) * EMB + k0 + ac];
    } else {
#pragma unroll
      for (int i = 0; i < 4; ++i)
        regb[i] = *(const float4*)&B[(size_t)(k0 + br) * EMB + blockN + bc + 32 * i];
    }
  };

  auto store_tile = [&](int buf) {
#pragma unroll
    for (int i = 0; i < 4; ++i)
      *(float4*)&As[buf][(ar + 32 * i) * SA + ac] = rega[i];
    if constexpr (BT) {
#pragma unroll
      for (int i = 0; i < 4; ++i)
        *(float4*)&Bs[buf][(ar + 32 * i) * SA + ac] = regb[i];
    } else {
#pragma unroll
      for (int i = 0; i < 4; ++i)
        *(float4*)&Bs[buf][br * SBN + bc + 32 * i] = regb[i];
    }
  };

  load_tile(0);
  store_tile(0);
  __syncthreads();

  for (int k0 = 0; k0 < EMB; k0 += BK) {
    const int cur = (k0 >> 4) & 1;
    const bool more = (k0 + BK) < EMB;
    if (more) load_tile(k0 + BK);  // overlap next-slab global loads with compute

    const float* as = &As[cur][0];
    const float* bs = &Bs[cur][0];
#pragma unroll
    for (int kk = 0; kk < BK; kk += 4) {
      const int kf = kk + lhalf * 2;  // this lane's K pair
      v2f af[4], bf[4];
#pragma unroll
      for (int mt = 0; mt < 4; ++mt)
        af[mt] = *(const v2f*)&as[(waveM + 16 * mt + lmod) * SA + kf];
#pragma unroll
      for (int nt = 0; nt < 4; ++nt) {
        if constexpr (BT) {
          bf[nt] = *(const v2f*)&bs[(waveN + 16 * nt + lmod) * SA + kf];
        } else {
          bf[nt].x = bs[kf * SBN + waveN + 16 * nt + lmod];
          bf[nt].y = bs[(kf + 1) * SBN + waveN + 16 * nt + lmod];
        }
      }
#pragma unroll
      for (int mt = 0; mt < 4; ++mt)
#pragma unroll
        for (int nt = 0; nt < 4; ++nt)
          acc[mt][nt] = __builtin_amdgcn_wmma_f32_16x16x4_f32(
              false, af[mt], false, bf[nt], (short)0, acc[mt][nt], false, false);
    }

    if (more) store_tile(cur ^ 1);
    __syncthreads();  // single barrier per K-slab (double buffer)
  }

  // Epilogue: C/D layout — VGPR r holds M = lhalf*8 + r, N = lmod
#pragma unroll
  for (int mt = 0; mt < 4; ++mt) {
#pragma unroll
    for (int nt = 0; nt < 4; ++nt) {
      const int col = blockN + waveN + 16 * nt + lmod;
#pragma unroll
      for (int r = 0; r < 8; ++r) {
        const int row = blockM + waveM + 16 * mt + lhalf * 8 + r;
        C[(size_t)row * EMB + col] = acc[mt][nt][r] * alpha;
      }
    }
  }
}

// In-place row softmax over [EMB, EMB], one 256-thread block per row.
__global__ __launch_bounds__(256) void softmax_rows(float* __restrict__ data) {
  float* p = data + (size_t)blockIdx.x * EMB;
  const int tid = threadIdx.x;
  __shared__ float redm[8];
  __shared__ float reds[8];

  float m = -3.402823466e38f;
  for (int i = tid; i < EMB; i += 256) m = fmaxf(m, p[i]);
#pragma unroll
  for (int off = 16; off; off >>= 1) m = fmaxf(m, __shfl_xor(m, off, 32));
  if ((tid & 31) == 0) redm[tid >> 5] = m;
  __syncthreads();
  float M = redm[0];
#pragma unroll
  for (int i = 1; i < 8; ++i) M = fmaxf(M, redm[i]);

  float s = 0.f;
  for (int i = tid; i < EMB; i += 256) {
    float v = __expf(p[i] - M);
    p[i] = v;
    s += v;
  }
#pragma unroll
  for (int off = 16; off; off >>= 1) s += __shfl_xor(s, off, 32);
  if ((tid & 31) == 0) reds[tid >> 5] = s;
  __syncthreads();
  float S = 0.f;
#pragma unroll
  for (int i = 0; i < 8; ++i) S += reds[i];
  const float inv = 1.0f / S;
  for (int i = tid; i < EMB; i += 256) p[i] *= inv;
}

extern "C" void kernel_launch(void* const* d_in, const int* in_sizes, int n_in,
                              void* d_out, int out_size, void* d_ws, size_t ws_size,
                              hipStream_t stream) {
  // inputs: [0]=tkns (dead — Q/K/V never reach the outputs), [1]=Wq, [2]=Wk, [3]=Wv
  const float* Wq = (const float*)d_in[1];
  const float* Wk = (const float*)d_in[2];
  const float* Wv = (const float*)d_in[3];
  float* attn = (float*)d_out;             // output 0: [EMB, EMB]
  float* sftm = attn + (size_t)EMB * EMB;  // output 1: [EMB, EMB]

  dim3 grid(EMB / BN, EMB / BM);
  dim3 block(128);

  // scores = (Wq @ Wk^T) * (1/sqrt(4096)) written into attn
  gemm_wmma_f32<true><<<grid, block, 0, stream>>>(Wq, Wk, attn, 0.015625f);
  // attn = softmax(scores) in place
  softmax_rows<<<EMB, 256, 0, stream>>>(attn);
  // sftmax = attn @ Wv
  gemm_wmma_f32<false><<<grid, block, 0, stream>>>(attn, Wv, sftm, 1.0f);
}